// RingNN_45457933861258
// MI455X (gfx1250) — compile-verified
//
#include <hip/hip_runtime.h>
#include <math.h>

typedef __attribute__((ext_vector_type(2))) float v2f;
typedef __attribute__((ext_vector_type(8))) float v8f;

#define THREADS 256
#define SPW 8              // samples per workgroup

// ---- workspace layout (floats) ----
#define WS_W1C 0           // 16: cos(conv1 w)
#define WS_W1S 16          // 16: sin(conv1 w)
#define WS_B2  32          // 128x16: conv2 B (cols 0-7 dir_x, 8-15 dir_y)
#define WS_BFX 2080        // 576x16: ff B for dir_x (cols 0-9 valid)
#define WS_BFY 11296       // 576x16: ff B for dir_y

// ============ stage 0: trig of all weights -> ws ============
__global__ void ring_prep(const float* __restrict__ w1,
                          const float* __restrict__ w2,
                          const float* __restrict__ ffw,
                          float* __restrict__ ws) {
  int tid = threadIdx.x;
  if (tid < 16) {
    float w = w1[tid];
    ws[WS_W1C + tid] = cosf(w);
    ws[WS_W1S + tid] = sinf(w);
  }
  // conv2 B: K=128 ([cosP|sinP] im2col), N=16 (8 x-cols, 8 y-cols)
  for (int i = tid; i < 128 * 16; i += THREADS) {
    int k = i >> 4, n = i & 15;
    int o = n & 7;
    int kk = (k < 64) ? k : (k - 64);          // k = c*16+kh*4+kw matches (8,4,4,4) flat
    float w = w2[o * 64 + kk];
    float v;
    if (n < 8) v = (k < 64) ?  cosf(w) : sinf(w);   // dir_x =  cosP*cosW + sinP*sinW
    else       v = (k < 64) ? -sinf(w) : cosf(w);   // dir_y = -cosP*sinW + sinP*cosW
    ws[WS_B2 + i] = v;
  }
  // ff B: K=576 ([cosH|sinH]), N=16 (cols 0-9 valid, rest 0)
  for (int i = tid; i < 576 * 16; i += THREADS) {
    int k = i >> 4, n = i & 15;
    float vx = 0.f, vy = 0.f;
    if (n < 10) {
      int kk = (k < 288) ? k : (k - 288);
      float w = ffw[kk * 10 + n];                   // ff_weight (288,10)
      if (k < 288) { vx = cosf(w); vy = -sinf(w); }
      else         { vx = sinf(w); vy =  cosf(w); }
    }
    ws[WS_BFX + i] = vx;
    ws[WS_BFY + i] = vy;
  }
}

// ============ fused network: 8 samples per workgroup ============
__global__ __launch_bounds__(THREADS)
void ring_fused(const float* __restrict__ x,
                const float* __restrict__ ws,
                float* __restrict__ out) {
  // LDS: regionA (12544 f) then regionB (4608 f) = 68608 B total.
  //  regionA: phase1-2 = A1 trig [cos: (b*4+c)*196+p | sin: +6272]
  //           phase3-4 = FFA 16x576 (rows 8-15 zero)   (aliased after barrier)
  //  regionB: phase2-3 = D2 [m*16+n], m = b*36+oh2*6+ow2
  //           phase4-5 = FF partials [wave*512 + t*256 + v*32 + lane], red @4096
  __shared__ float smem[17152];
  float* A1  = smem;
  float* FFA = smem;
  float* D2  = smem + 12544;
  float* FFP = smem + 12544;

  const int tid  = threadIdx.x;
  const int lane = tid & 31;
  // Wave index as an SGPR: tid>>5 is wave-uniform; readfirstlane lets the
  // compiler use scalar branches around the WMMA loops so EXEC stays all-1s
  // through every v_wmma (ISA requirement), instead of exec-masked vector loops.
  const int wave = __builtin_amdgcn_readfirstlane(tid >> 5);
  const int b0   = blockIdx.x * SPW;

  // ---- phase 1: conv1 (stride==kernel => each pixel read once, direct) ----
  float w1c[16], w1s[16];
  #pragma unroll
  for (int i = 0; i < 16; ++i) { w1c[i] = ws[WS_W1C + i]; w1s[i] = ws[WS_W1S + i]; }

  for (int i = tid; i < SPW * 196; i += THREADS) {
    int b = i / 196, p = i % 196;
    int oh = p / 14, ow = p % 14;
    const float* xb = x + (size_t)(b0 + b) * 784 + (oh * 2) * 28 + (ow * 2);
    float px[4] = { xb[0], xb[1], xb[28], xb[29] };
    float pc[4], ps[4];
    #pragma unroll
    for (int t = 0; t < 4; ++t) { pc[t] = cosf(px[t]); ps[t] = sinf(px[t]); }
    #pragma unroll
    for (int c = 0; c < 4; ++c) {
      float dx = 0.f, dy = 0.f;
      #pragma unroll
      for (int t = 0; t < 4; ++t) {
        float cw = w1c[c * 4 + t], sw = w1s[c * 4 + t];
        dx = fmaf(pc[t], cw, fmaf(ps[t],  sw, dx));
        dy = fmaf(ps[t], cw, fmaf(pc[t], -sw, dy));
      }
      float rinv = rsqrtf(fmaxf(dx * dx + dy * dy, 1e-30f));
      A1[(b * 4 + c) * 196 + p]        = dx * rinv;   // cos(theta1)
      A1[6272 + (b * 4 + c) * 196 + p] = dy * rinv;   // sin(theta1)
    }
  }
  __syncthreads();

  // ---- phase 2: conv2 as WMMA GEMM, M=288 K=128 N=16 (x|y) ----
  {
    const float* B2 = ws + WS_B2;
    const int n     = lane & 15;
    const int khalf = (lane >> 4) << 1;      // K offset 0 / 2 per lane half
    for (int t = wave; t < 18; t += 8) {     // 18 M-tiles over 8 waves (scalar loop)
      const int m0  = t * 16;
      const int row = m0 + (lane & 15);      // this lane's A row (M)
      const int b   = row / 36;
      const int pq  = row % 36;
      const int pbase = ((pq / 6) * 2) * 14 + (pq % 6) * 2;  // patch top-left in 14x14
      const int bbase = b * 784;
      auto aElem = [&](int kk) -> float {    // im2col gather from LDS
        int part = (kk >= 64) ? 6272 : 0;
        int kq = kk & 63;
        return A1[part + bbase + (kq >> 4) * 196 + pbase + ((kq >> 2) & 3) * 14 + (kq & 3)];
      };
      v8f acc = {};
      #pragma unroll 4
      for (int ks = 0; ks < 32; ++ks) {
        const int k0 = ks * 4 + khalf;
        v2f a, bm;
        a.x  = aElem(k0);
        a.y  = aElem(k0 + 1);
        bm.x = B2[k0 * 16 + n];
        bm.y = B2[(k0 + 1) * 16 + n];
        acc = __builtin_amdgcn_wmma_f32_16x16x4_f32(false, a, false, bm,
                                                    (short)0, acc, false, false);
      }
      const int mloc = (lane >> 4) << 3;     // D rows 0-7 / 8-15 per lane half
      #pragma unroll
      for (int v = 0; v < 8; ++v)
        D2[(m0 + mloc + v) * 16 + (lane & 15)] = acc[v];
    }
  }
  __syncthreads();

  // ---- phase 3: theta2 -> cos/sin via x/r, y/r; build FF A matrix ----
  for (int i = tid; i < SPW * 288; i += THREADS) {
    int b = i / 288, j = i % 288;                       // j = oh2*48 + ow2*8 + c
    int m = b * 36 + (j / 48) * 6 + ((j >> 3) % 6);
    int c = j & 7;
    float dx = D2[m * 16 + c];
    float dy = D2[m * 16 + c + 8];
    float rinv = rsqrtf(fmaxf(dx * dx + dy * dy, 1e-30f));
    FFA[b * 576 + j]       = dx * rinv;
    FFA[b * 576 + 288 + j] = dy * rinv;
  }
  for (int i = tid; i < 8 * 576; i += THREADS) FFA[8 * 576 + i] = 0.f;  // pad rows 8-15
  __syncthreads();

  // ---- phase 4: FF as WMMA GEMM, M=16 K=576, K split across 8 waves ----
  {
    const float* BFx = ws + WS_BFX;
    const float* BFy = ws + WS_BFY;
    const int n     = lane & 15;
    const int khalf = (lane >> 4) << 1;
    const int arow  = (lane & 15) * 576;
    v8f accX = {}, accY = {};
    for (int ks = wave; ks < 144; ks += 8) {           // scalar loop bounds
      const int k0 = ks * 4 + khalf;
      v2f a, bx, by;
      a.x  = FFA[arow + k0];
      a.y  = FFA[arow + k0 + 1];
      bx.x = BFx[k0 * 16 + n];  bx.y = BFx[(k0 + 1) * 16 + n];
      by.x = BFy[k0 * 16 + n];  by.y = BFy[(k0 + 1) * 16 + n];
      accX = __builtin_amdgcn_wmma_f32_16x16x4_f32(false, a, false, bx,
                                                   (short)0, accX, false, false);
      accY = __builtin_amdgcn_wmma_f32_16x16x4_f32(false, a, false, by,
                                                   (short)0, accY, false, false);
    }
    #pragma unroll
    for (int v = 0; v < 8; ++v) {
      FFP[wave * 512 +       v * 32 + lane] = accX[v];
      FFP[wave * 512 + 256 + v * 32 + lane] = accY[v];
    }
  }
  __syncthreads();

  // ---- phase 5: reduce wave partials, emit sin(atan2(y,x)) = y/r ----
  for (int s = tid; s < 512; s += THREADS) {
    float acc = 0.f;
    #pragma unroll
    for (int w = 0; w < 8; ++w) acc += FFP[w * 512 + s];
    FFP[4096 + s] = acc;
  }
  __syncthreads();
  for (int i = tid; i < SPW * 10; i += THREADS) {
    int b = i / 10, n10 = i % 10;        // sample b -> D row b (lane=n10, v=b)
    float dx = FFP[4096 +       b * 32 + n10];
    float dy = FFP[4096 + 256 + b * 32 + n10];
    float rinv = rsqrtf(fmaxf(dx * dx + dy * dy, 1e-30f));
    out[(size_t)(b0 + b) * 10 + n10] = dy * rinv;
  }
}

extern "C" void kernel_launch(void* const* d_in, const int* in_sizes, int n_in,
                              void* d_out, int out_size, void* d_ws, size_t ws_size,
                              hipStream_t stream) {
  (void)n_in; (void)out_size; (void)ws_size;
  const float* x   = (const float*)d_in[0];
  const float* w1  = (const float*)d_in[1];
  const float* w2  = (const float*)d_in[2];
  const float* ffw = (const float*)d_in[3];
  float* ws  = (float*)d_ws;
  float* out = (float*)d_out;

  ring_prep<<<1, THREADS, 0, stream>>>(w1, w2, ffw, ws);

  int nB = in_sizes[0] / 784;                         // 4096 samples
  int grid = (nB + SPW - 1) / SPW;                    // 512 workgroups
  ring_fused<<<grid, THREADS, 0, stream>>>(x, ws, out);
}